// Time_attention_60859686584687
// MI455X (gfx1250) — compile-verified
//
#include <hip/hip_runtime.h>

// ---------------------------------------------------------------------------
// Time-attention (B=16, CH=64, T=2048, D=1, F=8) for gfx1250 / MI455X.
//
// energy = a*(X^T X) + p*S_t + r*S_s + c   ->  minmax-norm  ->  softmax
// out    = gamma * (softmax @ X^T)^T + x
//
// Both T x T GEMMs (K = 64 and K = 2048) run on V_WMMA_F32_16X16X4_F32,
// keeping full fp32 precision. Each GEMM uses TWO independent accumulators
// (even/odd k-steps) so consecutive v_wmma issues have no RAW dependency
// (avoids the compiler's s_delay_alu VALU_DEP_1 stalls between chained WMMAs).
// ---------------------------------------------------------------------------

typedef __attribute__((ext_vector_type(2))) float v2f;
typedef __attribute__((ext_vector_type(8))) float v8f;

#define TT      2048
#define CHN     64
#define NBATCH  16
#define ROWS    16          // energy rows per workgroup
#define ESTR    2052        // padded LDS row stride (floats): 2052 % 64 = 4 banks/row
#define VSTR    65          // padded V-tile stride (floats)

__global__ __launch_bounds__(256) void ta_fused(
    const float* __restrict__ x,
    const float* __restrict__ w1, const float* __restrict__ b1,
    const float* __restrict__ w2, const float* __restrict__ b2,
    const float* __restrict__ gamma,
    float* __restrict__ out)
{
    __shared__ float Ebuf[ROWS * ESTR];   // 131,328 B : energy / exp row-block
    __shared__ float Vt[2][32 * VSTR];    //  16,640 B : double V tile (also out staging)
    __shared__ float invD[ROWS];          // softmax 1/denominator per row

    const int b     = blockIdx.y;
    const int tBase = blockIdx.x * ROWS;
    const int tid   = threadIdx.x;
    const int lane  = tid & 31;
    const int w     = tid >> 5;      // wave id 0..7
    const int half  = lane >> 4;     // 0: lanes 0-15, 1: lanes 16-31
    const int l15   = lane & 15;

    // ---- scalar coefficients from the 1x1 convs (F = 8) --------------------
    float a_s = 0.f, p_s = 0.f, r_s = 0.f, c_s = 0.f;
    for (int f = 0; f < 8; ++f) {
        float W1 = w1[f], W2 = w2[f], B1 = b1[f], B2 = b2[f];
        a_s += W1 * W2; p_s += W1 * B2; r_s += B1 * W2; c_s += B1 * B2;
    }
    c_s *= (float)CHN;

    const float* xb = x + (size_t)b * CHN * TT;

    // =========================================================================
    // Pass A:  E = a * (Q K^T) + p*S_t + r*S_s + c      (Q = K = X^T rows)
    // A operand (16 rows x 64 chans) lives in 32 VGPRs, reused for 16 tiles.
    // WMMA f32 16x16x4 A layout: lane = M (0..15), VGPR pair = K at 2*half.
    // =========================================================================
    v2f areg[16];
    {
        const int tRow = tBase + l15;
        for (int kk = 0; kk < 16; ++kk) {
            const int cA = 4 * kk + 2 * half;
            areg[kk].x = xb[(size_t)cA       * TT + tRow];
            areg[kk].y = xb[(size_t)(cA + 1) * TT + tRow];
        }
    }
    // channel-sum of the Q rows, extracted from areg (free): lane l15 holds S[tBase+l15]
    float sq = 0.f;
    for (int kk = 0; kk < 16; ++kk) sq += areg[kk].x + areg[kk].y;
    const float sQfull = sq + __shfl_xor(sq, 16);

    for (int i = 0; i < 16; ++i) {
        const int s0 = w * 256 + i * 16;           // this wave's column tile
        v8f acc0 = {}, acc1 = {};                  // independent even/odd chains
        float bs = 0.f;                            // column channel-sums (free)
        for (int kk = 0; kk < 16; kk += 2) {
            const int cB0 = 4 * kk + 2 * half;
            const int cB1 = 4 * (kk + 1) + 2 * half;
            v2f b0, b1v;
            b0.x  = xb[(size_t)cB0       * TT + s0 + l15];
            b0.y  = xb[(size_t)(cB0 + 1) * TT + s0 + l15];
            b1v.x = xb[(size_t)cB1       * TT + s0 + l15];
            b1v.y = xb[(size_t)(cB1 + 1) * TT + s0 + l15];
            bs += b0.x + b0.y + b1v.x + b1v.y;
            acc0 = __builtin_amdgcn_wmma_f32_16x16x4_f32(
                     false, areg[kk],     false, b0,  (short)0, acc0, false, false);
            acc1 = __builtin_amdgcn_wmma_f32_16x16x4_f32(
                     false, areg[kk + 1], false, b1v, (short)0, acc1, false, false);
        }
        const float Scol = bs + __shfl_xor(bs, 16);   // S[s0 + l15]
        for (int v = 0; v < 8; ++v) {
            const float Srow = __shfl(sQfull, v + 8 * half);   // S[row of acc[v]]
            const float e = a_s * (acc0[v] + acc1[v]) + p_s * Srow + r_s * Scol + c_s;
            Ebuf[(v + 8 * half) * ESTR + s0 + l15] = e;
        }
    }
    __syncthreads();

    // =========================================================================
    // Pass B: per-row min/max, min-max normalize, exp in-place, denominators.
    // softmax((e-min)/(max-min+eps)) == normalized exp((e-min)*scale).
    // 16 threads per row (aligned half-wave -> shfl_xor reductions).
    // =========================================================================
    {
        const int r   = tid >> 4;
        const int cst = tid & 15;
        float mn = 3.402823466e38f, mx = -3.402823466e38f;
        for (int j = cst; j < TT; j += 16) {
            const float e = Ebuf[r * ESTR + j];
            mn = fminf(mn, e); mx = fmaxf(mx, e);
        }
        for (int m = 1; m < 16; m <<= 1) {
            mn = fminf(mn, __shfl_xor(mn, m));
            mx = fmaxf(mx, __shfl_xor(mx, m));
        }
        const float scale = 1.0f / (mx - mn + 1e-8f);
        float sum = 0.f;
        for (int j = cst; j < TT; j += 16) {
            const float z = __expf((Ebuf[r * ESTR + j] - mn) * scale);
            Ebuf[r * ESTR + j] = z;
            sum += z;
        }
        for (int m = 1; m < 16; m <<= 1) sum += __shfl_xor(sum, m);
        if (cst == 0) invD[r] = 1.0f / sum;
    }
    __syncthreads();

    // =========================================================================
    // Pass C: out_raw(16 x 64) = expE(16 x 2048) @ X^T(2048 x 64), K split in 2
    // wave-groups (g = w>>2), 4 N-tiles each -> all 8 waves run WMMA.
    // V tiles staged coalesced into padded LDS; E read via bank-spread rows.
    // Two accumulators (even/odd kk) carried across chunks, merged at the end.
    // =========================================================================
    const int g   = w >> 2;            // K-half: 0 -> j in [0,1024), 1 -> [1024,2048)
    const int c0  = (w & 3) * 16;      // output channel tile
    v8f acc2a = {}, acc2b = {};
    for (int ic = 0; ic < 32; ++ic) {
        // stage 32 j-values for BOTH K-halves:  Vt[h][jj][c] = x[b, c, jBase+jj]
        for (int e = tid; e < 4096; e += 256) {
            const int h   = e >> 11;
            const int rem = e & 2047;
            const int c   = rem >> 5;
            const int jj  = rem & 31;
            Vt[h][jj * VSTR + c] = xb[(size_t)c * TT + h * 1024 + ic * 32 + jj];
        }
        __syncthreads();
        const int jAbs = g * 1024 + ic * 32;
        for (int kk = 0; kk < 8; kk += 2) {
            const int k0 = 4 * kk + 2 * half;
            const int k1 = 4 * (kk + 1) + 2 * half;
            v2f aE0, bV0, aE1, bV1;
            aE0.x = Ebuf[l15 * ESTR + jAbs + k0];
            aE0.y = Ebuf[l15 * ESTR + jAbs + k0 + 1];
            bV0.x = Vt[g][k0 * VSTR + c0 + l15];
            bV0.y = Vt[g][(k0 + 1) * VSTR + c0 + l15];
            aE1.x = Ebuf[l15 * ESTR + jAbs + k1];
            aE1.y = Ebuf[l15 * ESTR + jAbs + k1 + 1];
            bV1.x = Vt[g][k1 * VSTR + c0 + l15];
            bV1.y = Vt[g][(k1 + 1) * VSTR + c0 + l15];
            acc2a = __builtin_amdgcn_wmma_f32_16x16x4_f32(
                      false, aE0, false, bV0, (short)0, acc2a, false, false);
            acc2b = __builtin_amdgcn_wmma_f32_16x16x4_f32(
                      false, aE1, false, bV1, (short)0, acc2b, false, false);
        }
        __syncthreads();
    }

    // stage K-partials into LDS (reuse Vt), waves write disjoint [g][rows][c0..]
    for (int v = 0; v < 8; ++v)
        Vt[g][(v + 8 * half) * VSTR + c0 + l15] = acc2a[v] + acc2b[v];
    __syncthreads();

    // epilogue: out[b,c,t] = gamma * (part0+part1) * invD[t] + x[b,c,t]
    const float gm = gamma[0];
    for (int e = tid; e < ROWS * CHN; e += 256) {
        const int ttl = e & 15;        // t within block (inner -> coalesced)
        const int c   = e >> 4;
        const float m = (Vt[0][ttl * VSTR + c] + Vt[1][ttl * VSTR + c]) * invD[ttl] * gm;
        const size_t idx = (size_t)b * CHN * TT + (size_t)c * TT + tBase + ttl;
        out[idx] = m + x[idx];
    }
}

extern "C" void kernel_launch(void* const* d_in, const int* in_sizes, int n_in,
                              void* d_out, int out_size, void* d_ws, size_t ws_size,
                              hipStream_t stream) {
    (void)in_sizes; (void)n_in; (void)out_size; (void)d_ws; (void)ws_size;
    const float* x     = (const float*)d_in[0];
    const float* w1    = (const float*)d_in[1];
    const float* b1    = (const float*)d_in[2];
    const float* w2    = (const float*)d_in[3];
    const float* b2    = (const float*)d_in[4];
    const float* gamma = (const float*)d_in[5];
    float* out = (float*)d_out;

    dim3 grid(TT / ROWS, NBATCH);   // 128 row-blocks x 16 batches = 2048 WGs
    ta_fused<<<grid, 256, 0, stream>>>(x, w1, b1, w2, b2, gamma, out);
}